// GNN_19585050869930
// MI455X (gfx1250) — compile-verified
//
#include <hip/hip_runtime.h>

typedef float v2f __attribute__((ext_vector_type(2)));
typedef float v8f __attribute__((ext_vector_type(8)));

// ---------------- degree / normalization ----------------

__global__ __launch_bounds__(256) void k_degree(const int* __restrict__ edst, int E,
                                                float* __restrict__ deg) {
    int i = blockIdx.x * 256 + threadIdx.x;
    if (i < E) atomicAdd(&deg[edst[i]], 1.0f);
}

__global__ __launch_bounds__(256) void k_dinv(float* __restrict__ dinv, int n) {
    int i = blockIdx.x * 256 + threadIdx.x;
    if (i < n) dinv[i] = rsqrtf(dinv[i] + 1.0f);  // +1 for the self-loop
}

// ---------------- layer 1 GEMM: xw1 = x @ W1  (f32 WMMA 16x16x4) ----------------
// One wave per 16-row M-tile; W1 (20x64, 5KB) staged in LDS.

__global__ __launch_bounds__(256) void k_gemm1(const float* __restrict__ x,
                                               const float* __restrict__ W1,
                                               float* __restrict__ xw, int mtiles) {
    __shared__ float sW[20 * 64];
    for (int t = threadIdx.x; t < 20 * 64; t += 256) sW[t] = W1[t];
    __syncthreads();

    int lane  = threadIdx.x & 31;
    int mtile = blockIdx.x * 8 + (threadIdx.x >> 5);
    if (mtile >= mtiles) return;

    int m     = mtile * 16 + (lane & 15);
    int khalf = (lane >> 4) << 1;          // 0 (lanes 0-15) or 2 (lanes 16-31)
    int nbase = lane & 15;
    const float* xr = x + (size_t)m * 20;

    v2f a[5];
#pragma unroll
    for (int kk = 0; kk < 5; ++kk)         // K=20 -> 5 steps of 4
        a[kk] = *(const v2f*)(xr + 4 * kk + khalf);

    int row0 = mtile * 16 + ((lane >> 4) << 3);
#pragma unroll
    for (int nt = 0; nt < 4; ++nt) {       // N=64 -> 4 tiles of 16
        v8f c = {};
#pragma unroll
        for (int kk = 0; kk < 5; ++kk) {
            int krow = 4 * kk + khalf;
            v2f b = { sW[krow * 64 + nt * 16 + nbase],
                      sW[(krow + 1) * 64 + nt * 16 + nbase] };
            c = __builtin_amdgcn_wmma_f32_16x16x4_f32(false, a[kk], false, b,
                                                      (short)0, c, false, false);
        }
#pragma unroll
        for (int r = 0; r < 8; ++r)
            xw[(size_t)(row0 + r) * 64 + nt * 16 + nbase] = c[r];
    }
}

// ---------------- layer 1 aggregation ----------------
// init: agg = xw * dinv^2 (self-loop term, also serves as the zero-init)

__global__ __launch_bounds__(256) void k_selfinit1(const float* __restrict__ xw,
                                                   const float* __restrict__ dinv,
                                                   float* __restrict__ agg, int n) {
    int t = blockIdx.x * 256 + threadIdx.x;
    if (t < n * 64) {
        float di = dinv[t >> 6];
        agg[t] = xw[t] * di * di;
    }
}

// One wave per edge; each lane gathers a float2 and issues two f32 atomics.
__global__ __launch_bounds__(256) void k_scatter1(const int* __restrict__ esrc,
                                                  const int* __restrict__ edst, int E,
                                                  const float* __restrict__ dinv,
                                                  const float* __restrict__ xw,
                                                  float* __restrict__ agg) {
    long long tid = (long long)blockIdx.x * 256 + threadIdx.x;
    long long e   = tid >> 5;
    int lane      = (int)(tid & 31);
    if (e >= E) return;
    int s = esrc[e], d = edst[e];
    float nrm = dinv[s] * dinv[d];
    v2f v = *(const v2f*)(xw + (size_t)s * 64 + 2 * lane);
    float* p = agg + (size_t)d * 64 + 2 * lane;
    atomicAdd(p,     v.x * nrm);
    atomicAdd(p + 1, v.y * nrm);
}

__global__ __launch_bounds__(256) void k_relu_bias(float* __restrict__ agg,
                                                   const float* __restrict__ b1, int n) {
    int t = blockIdx.x * 256 + threadIdx.x;
    if (t < n * 64) {
        float v = agg[t] + b1[t & 63];
        agg[t] = v > 0.0f ? v : 0.0f;
    }
}

// ---------------- layer 2 GEMM: xw2 = h @ W2  (N padded 2 -> 16 in LDS) ----------------

__global__ __launch_bounds__(256) void k_gemm2(const float* __restrict__ h,
                                               const float* __restrict__ W2,
                                               float* __restrict__ xw2, int mtiles) {
    __shared__ float sW[64 * 16];
    for (int t = threadIdx.x; t < 64 * 16; t += 256) {
        int nn = t & 15;
        sW[t] = (nn < 2) ? W2[(t >> 4) * 2 + nn] : 0.0f;
    }
    __syncthreads();

    int lane  = threadIdx.x & 31;
    int mtile = blockIdx.x * 8 + (threadIdx.x >> 5);
    if (mtile >= mtiles) return;

    int m     = mtile * 16 + (lane & 15);
    int khalf = (lane >> 4) << 1;
    int nbase = lane & 15;
    const float* hr = h + (size_t)m * 64;

    v8f c = {};
#pragma unroll
    for (int kk = 0; kk < 16; ++kk) {      // K=64 -> 16 steps of 4
        int krow = 4 * kk + khalf;
        v2f a = *(const v2f*)(hr + krow);
        v2f b = { sW[krow * 16 + nbase], sW[(krow + 1) * 16 + nbase] };
        c = __builtin_amdgcn_wmma_f32_16x16x4_f32(false, a, false, b,
                                                  (short)0, c, false, false);
    }
    if (nbase < 2) {                        // only columns 0..1 are real
        int row0 = mtile * 16 + ((lane >> 4) << 3);
#pragma unroll
        for (int r = 0; r < 8; ++r)
            xw2[(size_t)(row0 + r) * 2 + nbase] = c[r];
    }
}

// ---------------- layer 2 aggregation ----------------

__global__ __launch_bounds__(256) void k_selfinit2(const float* __restrict__ xw2,
                                                   const float* __restrict__ dinv,
                                                   const float* __restrict__ b2,
                                                   float* __restrict__ out, int n) {
    int t = blockIdx.x * 256 + threadIdx.x;
    if (t < n * 2) {
        float di = dinv[t >> 1];
        out[t] = xw2[t] * di * di + b2[t & 1];
    }
}

__global__ __launch_bounds__(256) void k_scatter2(const int* __restrict__ esrc,
                                                  const int* __restrict__ edst, int E,
                                                  const float* __restrict__ dinv,
                                                  const float* __restrict__ xw2,
                                                  float* __restrict__ out) {
    int e = blockIdx.x * 256 + threadIdx.x;
    if (e >= E) return;
    int s = esrc[e], d = edst[e];
    float nrm = dinv[s] * dinv[d];
    v2f v = *(const v2f*)(xw2 + (size_t)s * 2);
    atomicAdd(&out[(size_t)d * 2],     v.x * nrm);
    atomicAdd(&out[(size_t)d * 2 + 1], v.y * nrm);
}

// ---------------- launcher ----------------

extern "C" void kernel_launch(void* const* d_in, const int* in_sizes, int n_in,
                              void* d_out, int out_size, void* d_ws, size_t ws_size,
                              hipStream_t stream) {
    const float* x  = (const float*)d_in[0];   // [n, 20]
    const int*   ei = (const int*)d_in[1];     // [2, E]
    const float* W1 = (const float*)d_in[2];   // [20, 64]
    const float* b1 = (const float*)d_in[3];   // [64]
    const float* W2 = (const float*)d_in[4];   // [64, 2]
    const float* b2 = (const float*)d_in[5];   // [2]
    float* out = (float*)d_out;                // [n, 2]

    const int n = in_sizes[0] / 20;
    const int E = in_sizes[1] / 2;
    const int* esrc = ei;
    const int* edst = ei + E;

    // workspace: dinv | xw1 | agg1/h | xw2   (~52.4 MB total)
    char* ws = (char*)d_ws;
    size_t offDinv = 0;
    size_t offXw1  = (((size_t)n * 4) + 255) / 256 * 256;
    size_t offAgg  = offXw1 + (size_t)n * 64 * 4;
    size_t offXw2  = offAgg + (size_t)n * 64 * 4;
    float* dinv = (float*)(ws + offDinv);
    float* xw1  = (float*)(ws + offXw1);
    float* agg1 = (float*)(ws + offAgg);
    float* xw2  = (float*)(ws + offXw2);

    const int mtiles = (n + 15) / 16;   // n=100000 -> 6250 exact

    hipMemsetAsync(dinv, 0, (size_t)n * sizeof(float), stream);
    k_degree   <<<(E + 255) / 256, 256, 0, stream>>>(edst, E, dinv);
    k_dinv     <<<(n + 255) / 256, 256, 0, stream>>>(dinv, n);

    k_gemm1    <<<(mtiles + 7) / 8, 256, 0, stream>>>(x, W1, xw1, mtiles);
    k_selfinit1<<<((n * 64) + 255) / 256, 256, 0, stream>>>(xw1, dinv, agg1, n);
    {
        long long th = (long long)E * 32;
        k_scatter1<<<(unsigned)((th + 255) / 256), 256, 0, stream>>>(esrc, edst, E, dinv, xw1, agg1);
    }
    k_relu_bias<<<((n * 64) + 255) / 256, 256, 0, stream>>>(agg1, b1, n);

    k_gemm2    <<<(mtiles + 7) / 8, 256, 0, stream>>>(agg1, W2, xw2, mtiles);
    k_selfinit2<<<((n * 2) + 255) / 256, 256, 0, stream>>>(xw2, dinv, b2, out, n);
    k_scatter2 <<<(E + 255) / 256, 256, 0, stream>>>(esrc, edst, E, dinv, xw2, out);
}